// VectorQuantizer_73761768342107
// MI455X (gfx1250) — compile-verified
//
#include <hip/hip_runtime.h>
#include <math.h>

// ---------------------------------------------------------------------------
// VQ forward for MI455X (gfx1250, wave32, WMMA).
// x:          [32, 128, 1024] f32   (flat token n = b*1024 + l, dim k -> x[b,k,l])
// embeddings: [8192, 128] f32
// Outputs (concat): out[4194304], quantized[4194304], loss, fit, prenorm,
//                   entropy, k_sum[1048576], k_elem[8192]
// ---------------------------------------------------------------------------

typedef __attribute__((ext_vector_type(2))) float v2f;
typedef __attribute__((ext_vector_type(8))) float v8f;

#define N_EMB   8192
#define EMB_DIM 128
#define N_TOK   32768
#define X_ELEMS 4194304   // 32*128*1024
#define TOK_TILE 32       // tokens per workgroup in the WMMA kernel
#define CODE_BLK 128      // 8 waves * 16 codes
#define SA2_STRIDE 48     // float2 units; 2*48 % 64 == 32 -> half-waves use disjoint banks
#define SB2_STRIDE 144    // float2 units; 2*144 % 64 == 32 -> conflict-free b64 reads

// ---------------- kernel 0: s2[j] = sum_k emb[j,k]^2 ----------------
__global__ void vq_s2_kernel(const float* __restrict__ emb, float* __restrict__ s2) {
    int j = blockIdx.x * blockDim.x + threadIdx.x;
    if (j < N_EMB) {
        float s = 0.f;
        for (int k = 0; k < EMB_DIM; ++k) { float e = emb[(size_t)j * EMB_DIM + k]; s += e * e; }
        s2[j] = s;
    }
}

// ---------------- kernel 1: fused GEMM (WMMA f32) + argmin ----------------
// grid = 1024 (token tiles of 32), block = 256 (8 waves), dynamic LDS (~96.5 KB).
// Each wave computes two 16x16 C tiles (tokens 0-15 / 16-31) sharing B fragments:
// per k-step: 3x ds_load_b64, 2x v_wmma_f32_16x16x4_f32.
__global__ void vq_argmin_kernel(const float* __restrict__ x,
                                 const float* __restrict__ emb,
                                 const float* __restrict__ s2,
                                 int*   __restrict__ idx_out,
                                 float* __restrict__ mind_out) {
    extern __shared__ char smem[];
    v2f*   pairA = (v2f*)smem;                          // 64 k-pairs x 48 (32 tokens + pad)
    v2f*   pairB = pairA + 64 * SA2_STRIDE;             // 64 k-pairs x 144 (128 codes + pad)
    float* sS    = (float*)(pairB + 64 * SB2_STRIDE);   // 128

    const int tid  = threadIdx.x;
    const int wave = tid >> 5;
    const int lane = tid & 31;
    const int hi   = lane >> 4;     // half-wave select (K offset of fragment)
    const int nn   = lane & 15;     // N column of C / M row of A
    const int token_base = blockIdx.x * TOK_TILE;
    const int bb = blockIdx.x >> 5;              // batch (1024 tokens / 32 per tile)
    const int l0 = (blockIdx.x & 31) * TOK_TILE;

    // Stage A: 32 tokens x 128 dims as K-pairs. flat[t][k] = x[bb, k, l0+t]
    for (int i = tid; i < TOK_TILE * 64; i += 256) {
        int t = i & 31, p = i >> 5;
        const float* xp = x + (size_t)bb * (EMB_DIM * 1024) + (size_t)(2 * p) * 1024 + l0 + t;
        v2f a; a.x = xp[0]; a.y = xp[1024];
        pairA[p * SA2_STRIDE + t] = a;
    }
    __syncthreads();

    float minv0[8], minv1[8];
    int   mini0[8], mini1[8];
#pragma unroll
    for (int r = 0; r < 8; ++r) {
        minv0[r] = 3.4e38f; mini0[r] = 0;
        minv1[r] = 3.4e38f; mini1[r] = 0;
    }

    const int jn = 16 * wave + nn;   // this lane's code column within the 128-code block

    for (int blk = 0; blk < N_EMB / CODE_BLK; ++blk) {
        const int c0 = blk * CODE_BLK;
        __syncthreads();   // protect pairB reuse
        // Stage B block as K-pairs: pairB[p][j] = emb[c0+j][2p..2p+1] (coalesced float2)
        for (int i = tid; i < CODE_BLK * 64; i += 256) {
            int j = i >> 6, p = i & 63;
            pairB[p * SB2_STRIDE + j] =
                *(const v2f*)(emb + (size_t)(c0 + j) * EMB_DIM + 2 * p);
        }
        if (tid < CODE_BLK) sS[tid] = s2[c0 + tid];
        __syncthreads();

        v8f acc0 = {0.f,0.f,0.f,0.f,0.f,0.f,0.f,0.f};
        v8f acc1 = {0.f,0.f,0.f,0.f,0.f,0.f,0.f,0.f};
        for (int k0 = 0; k0 < EMB_DIM; k0 += 4) {
            const int p = (k0 >> 1) + hi;                 // K-pair row for this half-wave
            v2f a0 = pairA[p * SA2_STRIDE + nn];          // tokens 0-15
            v2f a1 = pairA[p * SA2_STRIDE + 16 + nn];     // tokens 16-31
            v2f b  = pairB[p * SB2_STRIDE + jn];
            acc0 = __builtin_amdgcn_wmma_f32_16x16x4_f32(false, a0, false, b,
                                                         (short)0, acc0, false, false);
            acc1 = __builtin_amdgcn_wmma_f32_16x16x4_f32(false, a1, false, b,
                                                         (short)0, acc1, false, false);
        }

        const float s2v = sS[jn];
        const int   jg  = c0 + jn;
#pragma unroll
        for (int r = 0; r < 8; ++r) {
            float c0v = s2v - 2.f * acc0[r];           // dist - s1 (s1 added later)
            float c1v = s2v - 2.f * acc1[r];
            if (c0v < minv0[r]) { minv0[r] = c0v; mini0[r] = jg; }
            if (c1v < minv1[r]) { minv1[r] = c1v; mini1[r] = jg; }
        }
    }

    // Cross-lane / cross-wave reduction; reuse (dead) pairB region as scratch.
    __syncthreads();
    float* red_v = (float*)pairB;            // 8 waves * 32 tokens * 16 = 4096
    int*   red_i = (int*)(red_v + 4096);     // 4096
#pragma unroll
    for (int r = 0; r < 8; ++r) {
        int m = r + 8 * hi;
        red_v[wave * 512 + m * 16 + nn]        = minv0[r];
        red_i[wave * 512 + m * 16 + nn]        = mini0[r];
        red_v[wave * 512 + (16 + m) * 16 + nn] = minv1[r];
        red_i[wave * 512 + (16 + m) * 16 + nn] = mini1[r];
    }
    __syncthreads();
    if (tid < TOK_TILE) {
        const int m = tid;
        float best = 3.4e38f; int bi = 0x7fffffff;
        for (int w = 0; w < 8; ++w)
            for (int c = 0; c < 16; ++c) {
                float v  = red_v[w * 512 + m * 16 + c];
                int   i2 = red_i[w * 512 + m * 16 + c];
                if (v < best || (v == best && i2 < bi)) { best = v; bi = i2; }
            }
        float s1v = 0.f;
        for (int p = 0; p < 64; ++p) {
            v2f a = pairA[p * SA2_STRIDE + m];
            s1v += a.x * a.x + a.y * a.y;
        }
        idx_out[token_base + m]  = bi;
        mind_out[token_base + m] = s1v + best;
    }
}

// ---------------- block reduce helper (blockDim.x == 256) ----------------
__device__ __forceinline__ float block_reduce_sum256(float v, float* red, int tid) {
    red[tid] = v;
    __syncthreads();
    for (int off = 128; off > 0; off >>= 1) {
        if (tid < off) red[tid] += red[tid + off];
        __syncthreads();
    }
    float r = red[0];
    __syncthreads();
    return r;
}

// ---------------- kernel 2: sum(x), sum(x^2) partials (prenorm) ----------------
__global__ void vq_xsum_kernel(const float* __restrict__ x,
                               float* __restrict__ psum, float* __restrict__ psum2) {
    __shared__ float red[256];
    const int tid = threadIdx.x;
    const int base = blockIdx.x * 256 + tid;
    float s = 0.f, sq = 0.f;
    for (int i = 0; i < 8; ++i) {
        float v = x[base + i * 524288];
        s += v; sq += v * v;
    }
    float ts = block_reduce_sum256(s, red, tid);
    float tq = block_reduce_sum256(sq, red, tid);
    if (tid == 0) { psum[blockIdx.x] = ts; psum2[blockIdx.x] = tq; }
}

// ---------------- kernel 3: gather -> out/quantized + commit-loss partials ----------------
// grid = 2048 (token tiles of 16), block = 256.
__global__ void vq_gather_kernel(const float* __restrict__ x,
                                 const float* __restrict__ emb,
                                 const int* __restrict__ idx,
                                 float* __restrict__ out,
                                 float* __restrict__ quant,
                                 float* __restrict__ ploss) {
    __shared__ int sJ[16];
    __shared__ float red[256];
    const int tid = threadIdx.x;
    const int token_base = blockIdx.x * 16;
    const int bb = blockIdx.x >> 6;
    const int l0 = (blockIdx.x & 63) * 16;
    if (tid < 16) sJ[tid] = idx[token_base + tid];
    __syncthreads();
    float acc = 0.f;
    for (int i = tid; i < 16 * EMB_DIM; i += 256) {
        int t = i & 15, k = i >> 4;
        int j = sJ[t];
        float q  = emb[(size_t)j * EMB_DIM + k];
        size_t off = (size_t)bb * (EMB_DIM * 1024) + (size_t)k * 1024 + l0 + t;
        float xv = x[off];
        float d  = q - xv;
        acc += d * d;
        quant[off] = q;
        out[off]   = xv + d;   // straight-through, computed like the reference
    }
    float tot = block_reduce_sum256(acc, red, tid);
    if (tid == 0) ploss[blockIdx.x] = tot;
}

// ---------------- kernel 4: deterministic segment-sum (k_sum, k_elem) ----------------
// 32 codes per block; 128 threads = one dim each; per-code accumulators in LDS
// (each thread owns accS[d][tid] -> single-owner RMW, deterministic n-order).
__global__ void vq_ksum_kernel(const float* __restrict__ x,
                               const int* __restrict__ idx,
                               float* __restrict__ ksum,
                               float* __restrict__ kelem) {
    __shared__ int   sIdx[2048];
    __shared__ float accS[32 * 128];
    __shared__ int   cntS[32];
    const int j0  = blockIdx.x * 32;
    const int tid = threadIdx.x;   // dim k
    for (int d = 0; d < 32; ++d) accS[d * 128 + tid] = 0.f;
    if (tid < 32) cntS[tid] = 0;
    __syncthreads();
    for (int c0 = 0; c0 < N_TOK; c0 += 2048) {
        for (int u = tid; u < 2048; u += 128) sIdx[u] = idx[c0 + u];
        __syncthreads();
        for (int u = 0; u < 2048; ++u) {
            int d = sIdx[u] - j0;
            if ((unsigned)d < 32u) {
                int n = c0 + u;
                accS[d * 128 + tid] +=
                    x[(size_t)(n >> 10) * (EMB_DIM * 1024) + (size_t)tid * 1024 + (n & 1023)];
                if (tid == 0) cntS[d]++;
            }
        }
        __syncthreads();
    }
    for (int d = 0; d < 32; ++d)
        ksum[(size_t)(j0 + d) * EMB_DIM + tid] = accS[d * 128 + tid];
    if (tid < 32) kelem[j0 + tid] = (float)cntS[tid];
}

// ---------------- kernel 5: scalars (loss, fit, prenorm, entropy) ----------------
__global__ void vq_finalize_kernel(const float* __restrict__ mind,
                                   const float* __restrict__ ploss,
                                   const float* __restrict__ psum,
                                   const float* __restrict__ psum2,
                                   const float* __restrict__ kelem,
                                   float* __restrict__ scal) {
    __shared__ float red[256];
    const int tid = threadIdx.x;
    // fit = mean(nan_to_num(min_distance))
    float s = 0.f;
    for (int i = tid; i < N_TOK; i += 256) { float v = mind[i]; s += (v != v) ? 0.f : v; }
    float fit = block_reduce_sum256(s, red, tid) / (float)N_TOK;
    // commit loss
    s = 0.f;
    for (int i = tid; i < 2048; i += 256) s += ploss[i];
    float loss = 0.25f * block_reduce_sum256(s, red, tid) / (float)X_ELEMS;
    // prenorm
    s = 0.f;
    for (int i = tid; i < 2048; i += 256) s += psum[i];
    float Sx = block_reduce_sum256(s, red, tid);
    s = 0.f;
    for (int i = tid; i < 2048; i += 256) s += psum2[i];
    float Sx2 = block_reduce_sum256(s, red, tid);
    float varS = (Sx2 - Sx * Sx / (float)X_ELEMS) / (float)X_ELEMS;
    float prenorm = sqrtf(varS > 0.f ? varS : 0.f);
    // entropy
    s = 0.f;
    for (int i = tid; i < N_EMB; i += 256) s += kelem[i];
    float T = block_reduce_sum256(s, red, tid);
    s = 0.f;
    for (int i = tid; i < N_EMB; i += 256) {
        float p = kelem[i] / T;
        s += -p * logf(p + 1e-8f);
    }
    float ent = block_reduce_sum256(s, red, tid);
    if (tid == 0) {
        scal[0] = loss;
        scal[1] = fit;
        scal[2] = prenorm;
        scal[3] = ent;
    }
}

// ---------------------------------------------------------------------------
extern "C" void kernel_launch(void* const* d_in, const int* in_sizes, int n_in,
                              void* d_out, int out_size, void* d_ws, size_t ws_size,
                              hipStream_t stream) {
    (void)in_sizes; (void)n_in; (void)out_size; (void)ws_size;
    const float* x   = (const float*)d_in[0];
    const float* emb = (const float*)d_in[1];

    float* out   = (float*)d_out;                   // [4194304]
    float* quant = out + X_ELEMS;                   // [4194304]
    float* scal  = quant + X_ELEMS;                 // loss, fit, prenorm, entropy
    float* ksum  = scal + 4;                        // [8192*128]
    float* kelem = ksum + (size_t)N_EMB * EMB_DIM;  // [8192]

    int*   ws_idx   = (int*)d_ws;              // [32768]
    float* ws_mind  = (float*)d_ws + N_TOK;    // [32768]
    float* ws_s2    = ws_mind + N_TOK;         // [8192]
    float* ws_psum  = ws_s2 + N_EMB;           // [2048]
    float* ws_psum2 = ws_psum + 2048;          // [2048]
    float* ws_ploss = ws_psum2 + 2048;         // [2048]

    vq_s2_kernel<<<N_EMB / 256, 256, 0, stream>>>(emb, ws_s2);

    const size_t lds_bytes =
        (size_t)(64 * SA2_STRIDE + 64 * SB2_STRIDE) * sizeof(v2f) +
        CODE_BLK * sizeof(float);   // 24576 + 73728 + 512 = 98,816 B
    vq_argmin_kernel<<<N_TOK / TOK_TILE, 256, lds_bytes, stream>>>(
        x, emb, ws_s2, ws_idx, ws_mind);

    vq_xsum_kernel<<<2048, 256, 0, stream>>>(x, ws_psum, ws_psum2);

    vq_gather_kernel<<<N_TOK / 16, 256, 0, stream>>>(
        x, emb, ws_idx, out, quant, ws_ploss);

    vq_ksum_kernel<<<N_EMB / 32, 128, 0, stream>>>(x, ws_idx, ksum, kelem);

    vq_finalize_kernel<<<1, 256, 0, stream>>>(
        ws_mind, ws_ploss, ws_psum, ws_psum2, kelem, scal);
}